// ClusterSeparationOptimizer_62543313764770
// MI455X (gfx1250) — compile-verified
//
#include <hip/hip_runtime.h>

typedef __attribute__((ext_vector_type(2))) float v2f;
typedef __attribute__((ext_vector_type(8))) float v8f;

#define CC 24
#define NN 1536
#define HH 40
#define HE 48                 // edge columns padded to 3 WMMA tiles of 16
#define MT (NN / 16)          // 96 point tiles per cluster
#define PAIRS (CC * CC)       // 576
#define TOTAL_WAVES (PAIRS * MT)  // 55296
#define EPS_REF 1e-8f
#define SEP_W 1.0f
#define T_PEN 0.1f
#define R_PEN 1.0f

__device__ __forceinline__ void xform(float x, float y, float mx, float my,
                                      float co, float si, float tx, float ty,
                                      float& ox, float& oy) {
  // centered·R + (medoid + trans), R = [[c, s], [-s, c]]
  float cx = x - mx, cy = y - my;
  ox = cx * co - cy * si + mx + tx;
  oy = cx * si + cy * co + my + ty;
}

// ---------------- Stage 1a: transform all points ----------------
__global__ void xform_pts_kernel(const float* __restrict__ pts,
                                 const float* __restrict__ med,
                                 const float* __restrict__ ang,
                                 const float* __restrict__ trans,
                                 float* __restrict__ ptsT) {
  int idx = blockIdx.x * blockDim.x + threadIdx.x;
  if (idx >= CC * NN) return;
  int c = idx / NN;
  float a = ang[c], co = cosf(a), si = sinf(a);
  float ox, oy;
  xform(pts[idx * 2 + 0], pts[idx * 2 + 1],
        med[c * 2 + 0], med[c * 2 + 1], co, si,
        trans[c * 2 + 0], trans[c * 2 + 1], ox, oy);
  ptsT[idx * 2 + 0] = ox;
  ptsT[idx * 2 + 1] = oy;
}

// ------- Stage 1b: transform hulls, build B fragments + validity -------
// edgeB[j][e] = { ex/elen, -ey/elen, -c0/elen, valid }
__global__ void build_edges_kernel(const float* __restrict__ hulls,
                                   const float* __restrict__ med,
                                   const float* __restrict__ ang,
                                   const float* __restrict__ trans,
                                   const unsigned char* __restrict__ hmask,
                                   float* __restrict__ edgeB,
                                   int* __restrict__ hull_ok) {
  int j = blockIdx.x;
  int e = threadIdx.x;
  float a = ang[j], co = cosf(a), si = sinf(a);
  float mx = med[j * 2 + 0], my = med[j * 2 + 1];
  float tx = trans[j * 2 + 0], ty = trans[j * 2 + 1];
  if (e < HE) {
    float bx = 0.f, by = 0.f, bc = 0.f, vf = 0.f;
    if (e < HH) {
      int e2 = (e + 1) % HH;  // jnp.roll(-1)
      float p1x, p1y, p2x, p2y;
      xform(hulls[(j * HH + e) * 2 + 0], hulls[(j * HH + e) * 2 + 1],
            mx, my, co, si, tx, ty, p1x, p1y);
      xform(hulls[(j * HH + e2) * 2 + 0], hulls[(j * HH + e2) * 2 + 1],
            mx, my, co, si, tx, ty, p2x, p2y);
      float ex = p2x - p1x, ey = p2y - p1y;
      float elen_raw = sqrtf(ex * ex + ey * ey);
      float inv = 1.0f / (elen_raw + EPS_REF);
      float c0 = ex * p1y - ey * p1x;
      bx = ex * inv;
      by = -ey * inv;
      bc = -c0 * inv;
      vf = (elen_raw > 1e-6f) ? 1.0f : 0.0f;
    }
    float* o = edgeB + (j * HE + e) * 4;
    o[0] = bx; o[1] = by; o[2] = bc; o[3] = vf;
  }
  if (e == 0) {
    int cnt = 0;
    for (int k = 0; k < HH; k++) cnt += hmask[j * HH + k] ? 1 : 0;
    hull_ok[j] = (cnt >= 3) ? 1 : 0;
  }
}

// ---------------- Stage 2: WMMA pair kernel ----------------
// One wave = one (pair, 16-point tile); 3x V_WMMA_F32_16X16X4_F32 over the
// 48 padded edge columns, then a shfl_xor butterfly edge-reduction.
__global__ __launch_bounds__(256) void sep_kernel(
    const float* __restrict__ ptsT, const float* __restrict__ edgeB,
    const int* __restrict__ hull_ok, const unsigned char* __restrict__ pmask,
    float* __restrict__ partials) {
  int wid = blockIdx.x * 8 + (threadIdx.x >> 5);
  int lane = threadIdx.x & 31;
  int pair = wid / MT;
  int mtile = wid - pair * MT;
  int i = pair / CC;       // points cluster
  int j = pair - i * CC;   // hull cluster
  // wave-uniform predicates -> EXEC stays all-1s around WMMA
  if (i == j || !hull_ok[j]) {
    if (lane == 0) partials[wid] = 0.0f;
    return;
  }
  int hl = lane & 15;
  // A fragment: row M=hl, K0=py K1=px (lanes 0-15), K2=1 K3=0 (lanes 16-31)
  v2f afrag;
  if (lane < 16) {
    int prow = mtile * 16 + hl;
    afrag.x = ptsT[(i * NN + prow) * 2 + 1];  // py
    afrag.y = ptsT[(i * NN + prow) * 2 + 0];  // px
  } else {
    afrag.x = 1.0f;
    afrag.y = 0.0f;
  }

  unsigned pos8 = 0xFFu, neg8 = 0xFFu;
  float mina[8];
#pragma unroll
  for (int v = 0; v < 8; v++) mina[v] = __builtin_inff();

#pragma unroll
  for (int et = 0; et < 3; et++) {
    int e0 = et * 16 + hl;
    const float* eb = edgeB + (j * HE + e0) * 4;
    // B fragment: K0=bx K1=by (lanes 0-15), K2=bc K3=0 (lanes 16-31)
    v2f bfrag;
    if (lane < 16) { bfrag.x = eb[0]; bfrag.y = eb[1]; }
    else           { bfrag.x = eb[2]; bfrag.y = 0.0f;  }
    float vflag = eb[3];

    v8f cz = {0.f, 0.f, 0.f, 0.f, 0.f, 0.f, 0.f, 0.f};
    v8f d = __builtin_amdgcn_wmma_f32_16x16x4_f32(
        false, afrag, false, bfrag, (short)0, cz, false, false);

    if (vflag > 0.5f) {  // edge column valid for this lane
#pragma unroll
      for (int v = 0; v < 8; v++) {
        float dv = d[v];
        if (!(dv >= -EPS_REF)) pos8 &= ~(1u << v);
        if (!(dv <= EPS_REF))  neg8 &= ~(1u << v);
        mina[v] = fminf(mina[v], fabsf(dv));
      }
    }
  }

  // Edge reduction across the 16 columns held by each lane group of 16.
  unsigned bits = pos8 | (neg8 << 8);
#pragma unroll
  for (int m = 1; m < 16; m <<= 1) bits &= __shfl_xor(bits, m, 32);
  float mr[8];
#pragma unroll
  for (int v = 0; v < 8; v++) {
    float x = mina[v];
#pragma unroll
    for (int m = 1; m < 16; m <<= 1) x = fminf(x, __shfl_xor(x, m, 32));
    mr[v] = x;
  }

  // Rows: lanes 0-15 hold M=v, lanes 16-31 hold M=v+8 (dup across hl).
  float sum = 0.0f;
  if (hl == 0) {
    int mbase = (lane >> 4) * 8;
#pragma unroll
    for (int v = 0; v < 8; v++) {
      bool inside = (((bits >> v) & 1u) | ((bits >> (8 + v)) & 1u)) != 0;
      if (inside) {
        int prow = mtile * 16 + mbase + v;
        if (pmask[i * NN + prow]) sum += 1.0f / (1.0f + expf(-mr[v]));
      }
    }
  }
  sum += __shfl_xor(sum, 16, 32);
  if (lane == 0) partials[wid] = sum;
}

// ---------------- Stage 3: deterministic reduction ----------------
__global__ void reduce_kernel(const float* __restrict__ partials,
                              const float* __restrict__ ang,
                              const float* __restrict__ trans,
                              float* __restrict__ out) {
  __shared__ float sm[256];
  int t = threadIdx.x;
  float s = 0.0f;
  for (int k = t; k < TOTAL_WAVES; k += 256) s += partials[k];
  sm[t] = s;
  __syncthreads();
  for (int w = 128; w > 0; w >>= 1) {
    if (t < w) sm[t] += sm[t + w];
    __syncthreads();
  }
  if (t == 0) {
    float pen = 0.0f;
    for (int c = 0; c < CC; c++) {
      pen += T_PEN * (trans[2 * c] * trans[2 * c] +
                      trans[2 * c + 1] * trans[2 * c + 1]);
      pen += R_PEN * ang[c] * ang[c];
    }
    out[0] = SEP_W * sm[0] + pen;
  }
}

extern "C" void kernel_launch(void* const* d_in, const int* in_sizes, int n_in,
                              void* d_out, int out_size, void* d_ws,
                              size_t ws_size, hipStream_t stream) {
  const float* pts   = (const float*)d_in[0];  // (C,N,2)
  const float* hulls = (const float*)d_in[1];  // (C,H,2)
  const float* med   = (const float*)d_in[2];  // (C,2)
  const float* ang   = (const float*)d_in[3];  // (C,)
  const float* trans = (const float*)d_in[4];  // (C,2)
  const unsigned char* pmask = (const unsigned char*)d_in[5];  // (C,N) bool
  const unsigned char* hmask = (const unsigned char*)d_in[6];  // (C,H) bool
  float* out = (float*)d_out;

  // workspace layout (floats): ptsT | edgeB | partials | hull_ok(int)
  float* ptsT = (float*)d_ws;                 // C*N*2   = 73728
  float* edgeB = ptsT + CC * NN * 2;          // C*HE*4  = 4608
  float* partials = edgeB + CC * HE * 4;      // 55296
  int* hullok = (int*)(partials + TOTAL_WAVES);

  xform_pts_kernel<<<(CC * NN + 255) / 256, 256, 0, stream>>>(pts, med, ang,
                                                              trans, ptsT);
  build_edges_kernel<<<CC, 64, 0, stream>>>(hulls, med, ang, trans, hmask,
                                            edgeB, hullok);
  sep_kernel<<<TOTAL_WAVES / 8, 256, 0, stream>>>(ptsT, edgeB, hullok, pmask,
                                                  partials);
  reduce_kernel<<<1, 256, 0, stream>>>(partials, ang, trans, out);
}